// Patcher_78280073937146
// MI455X (gfx1250) — compile-verified
//
#include <hip/hip_runtime.h>

// Problem constants (fixed by setup_inputs in the reference)
#define B_    8
#define C_    3
#define H_    1024
#define W_    1024
#define P_    24          // patch_size
#define S_    16          // stride
#define M_    4           // (P - S)/2 reflect pad
#define NH_   64
#define NW_   64
#define TPB_  4                                    // patches per block along W
#define NTHREADS 288                               // 12 * 24 (9 waves of 32)
#define OUT_PER_BLOCK (TPB_ * P_ * P_ * C_)        // 6912 floats = 27648 B
#define TOTAL_OUT (B_ * NH_ * NW_ * P_ * P_ * C_)  // 56,623,104

// reflect into [0,1024) for j in [-4, 1027] (jnp.pad mode='reflect', no edge dup)
__device__ __forceinline__ int reflect1024(int j) {
    j = (j < 0) ? -j : j;
    j = (j >= H_) ? (2 * H_ - 2 - j) : j;
    return j;
}

// CDNA5 async global->LDS copy (ASYNCcnt). lds_off = workgroup-relative LDS byte offset.
__device__ __forceinline__ void async_load_b32(unsigned lds_off, const float* gptr) {
    asm volatile("global_load_async_to_lds_b32 %0, %1, off"
                 :: "v"(lds_off), "v"((unsigned long long)(const void*)gptr)
                 : "memory");
}

// CDNA5 async LDS->global store, 16 B per lane (ASYNCcnt).
__device__ __forceinline__ void async_store_b128(const void* gptr, unsigned lds_off) {
    asm volatile("global_store_async_from_lds_b128 %0, %1, off"
                 :: "v"((unsigned long long)gptr), "v"(lds_off)
                 : "memory");
}

__device__ __forceinline__ void wait_async0() {
    asm volatile("s_wait_asynccnt 0x0" ::: "memory");
}

__global__ __launch_bounds__(NTHREADS) void patchify_kernel(const float* __restrict__ x,
                                                            float* __restrict__ out) {
    // Tile staged in OUTPUT layout: [p][pr][pc][c] -> linear == out linear order
    __shared__ __align__(16) float obuf[OUT_PER_BLOCK];

    const int bid = blockIdx.x;          // 0 .. 8191
    const int b   = bid >> 10;           // / (NH_ * 16)
    const int r   = bid & 1023;
    const int ih  = r >> 4;
    const int g   = r & 15;
    const int iw0 = g * TPB_;
    const int tid = threadIdx.x;

    // one divmod per THREAD (not per element): 288 = 12 rows x 24 cols
    const int pr0 = tid / P_;            // 0..11
    const int pc  = tid - pr0 * P_;      // 0..23

    const int h_base = ih * S_ - M_;
    const int w_base = iw0 * S_ - M_;
    const float* xb  = x + (size_t)b * (size_t)(C_ * H_ * W_);

    const unsigned lds0 =
        (unsigned)(unsigned long long)(const void*)&obuf[0];

    // ---- gather phase: global -> LDS (output-ordered) via async b32 copies ----
    #pragma unroll
    for (int p = 0; p < TPB_; ++p) {
        const int w = reflect1024(w_base + p * S_ + pc);
        #pragma unroll
        for (int half = 0; half < 2; ++half) {
            const int pr = pr0 + half * 12;
            const int h  = reflect1024(h_base + pr);
            const float* src = xb + (size_t)h * W_ + (size_t)w;
            // byte offset of (p, pr, pc, c=0) in output-linear tile
            const unsigned l0 =
                lds0 + (unsigned)(((p * (P_ * P_)) + pr * P_ + pc) * C_) * 4u;
            async_load_b32(l0 + 0u, src);                     // c = 0
            async_load_b32(l0 + 4u, src + (H_ * W_));         // c = 1
            async_load_b32(l0 + 8u, src + 2 * (H_ * W_));     // c = 2
        }
    }
    wait_async0();        // this wave's async copies landed in LDS
    __syncthreads();      // cross-wave visibility of the whole tile

    // ---- scatter phase: LDS -> global, linear, async b128 (16 B / lane) ----
    const size_t out_base =
        (size_t)((b * NH_ + ih) * NW_ + iw0) * (size_t)(P_ * P_ * C_);
    const char* dst0 = (const char*)(out + out_base);
    #pragma unroll
    for (int j = 0; j < OUT_PER_BLOCK / (4 * NTHREADS); ++j) {  // 6 iterations
        const unsigned k = (unsigned)(j * NTHREADS + tid) * 16u; // 16B chunk index
        async_store_b128(dst0 + k, lds0 + k);
    }
    wait_async0();        // stores drained before wave exit (ENDPGM also waits)
}

__global__ void write_dims_kernel(float* out) {
    if (threadIdx.x == 0) {
        out[TOTAL_OUT + 0] = 64.0f;  // nH
        out[TOTAL_OUT + 1] = 64.0f;  // nW
    }
}

extern "C" void kernel_launch(void* const* d_in, const int* in_sizes, int n_in,
                              void* d_out, int out_size, void* d_ws, size_t ws_size,
                              hipStream_t stream) {
    (void)in_sizes; (void)n_in; (void)d_ws; (void)ws_size;
    const float* x = (const float*)d_in[0];
    float* out = (float*)d_out;

    const int nblocks = B_ * NH_ * (NW_ / TPB_);   // 8192
    patchify_kernel<<<nblocks, NTHREADS, 0, stream>>>(x, out);

    if (out_size >= TOTAL_OUT + 2) {
        write_dims_kernel<<<1, 1, 0, stream>>>(out);
    }
}